// HGTLayer_47562467835950
// MI455X (gfx1250) — compile-verified
//
#include <hip/hip_runtime.h>
#include <math.h>

#define NN 100000
#define EE 600000
#define CC 128
#define HH 8
#define DD 16
#define NC (NN * CC)      /* 12,800,000 */
#define NH2 (NN * HH * 2) /* 1,600,000  */
#define EH (EE * HH)      /* 4,800,000  */
#define TT 5              /* 16-node M-tiles per block (80 nodes) */
#define TNODES (TT * 16)

typedef __attribute__((ext_vector_type(2))) float v2f;
typedef __attribute__((ext_vector_type(8))) float v8f;

__device__ __forceinline__ v8f wmma_f32(v2f a, v2f b, v8f c) {
  // D = A(16x4) * B(4x16) + C(16x16), all f32
  return __builtin_amdgcn_wmma_f32_16x16x4_f32(false, a, false, b, (short)0, c,
                                               false, false);
}

// ---------------------------------------------------------------------------
// Kernel 0: init scratch (attn accumulator = 0, segment max = -inf, den = 0)
// ---------------------------------------------------------------------------
__global__ __launch_bounds__(256) void k_init(float* __restrict__ attn,
                                              float* __restrict__ mx,
                                              float* __restrict__ den) {
  int i = blockIdx.x * 256 + threadIdx.x;
  int stride = gridDim.x * 256;
  for (int j = i; j < NC; j += stride) attn[j] = 0.0f;
  for (int j = i; j < NH2; j += stride) {
    mx[j] = -INFINITY;
    den[j] = 0.0f;
  }
}

// ---------------------------------------------------------------------------
// Kernel 1: fused  q = xWq+bq ; k = xWk+bk ; v = xWv+bv ;
//           kr[r] = per-head (k @ a_rel[r,h]) ; vr[r] = per-head (v @ m_rel[r,h])
// One block = 80 nodes (5 M-tiles), wave w = head w. Each B-fragment load
// feeds 15 WMMAs (3 matrices x 5 M-tiles) -> 5x less weight traffic and
// 15 independent accumulator chains to hide load latency.
// ---------------------------------------------------------------------------
__global__ __launch_bounds__(256) void k_qkv(
    const float* __restrict__ x, const float* __restrict__ Wk,
    const float* __restrict__ bk, const float* __restrict__ Wq,
    const float* __restrict__ bq, const float* __restrict__ Wv,
    const float* __restrict__ bv, const float* __restrict__ a_rel,
    const float* __restrict__ m_rel, float* __restrict__ q,
    float* __restrict__ kr, float* __restrict__ vr) {
  __shared__ float xs[TNODES * 128];  // 40 KB x tile, shared by all 8 waves
  __shared__ float tb[8][16 * 17];    // per-wave k/v tile staging (pad 17)

  const int tid = threadIdx.x;
  const int lane = tid & 31;
  const int h = tid >> 5;  // wave id == head id
  const int node0 = blockIdx.x * TNODES;

  // stage x tile (80 nodes x 128 feats) into LDS: 2560 float4 / 256 thr
  {
    const float4* xg = (const float4*)(x + node0 * 128);
    float4* xl = (float4*)xs;
    #pragma unroll
    for (int i = 0; i < (TNODES * 128 / 4) / 256; ++i)
      xl[tid + 256 * i] = xg[tid + 256 * i];
  }
  __syncthreads();

  const int col = lane & 15;  // also the A-operand row (M) index
  const int hi = lane >> 4;

  v8f cq[TT] = {};
  v8f ck[TT] = {};
  v8f cv[TT] = {};

  for (int k0 = 0; k0 < 128; k0 += 4) {
    const int ka = k0 + 2 * hi;
    const int wb = ka * 128 + h * 16 + col;
    v2f bqf, bkf, bvf;
    bqf.x = Wq[wb];
    bqf.y = Wq[wb + 128];
    bkf.x = Wk[wb];
    bkf.y = Wk[wb + 128];
    bvf.x = Wv[wb];
    bvf.y = Wv[wb + 128];
    #pragma unroll
    for (int t = 0; t < TT; ++t) {
      v2f a;
      a.x = xs[(t * 16 + col) * 128 + ka];
      a.y = xs[(t * 16 + col) * 128 + ka + 1];
      cq[t] = wmma_f32(a, bqf, cq[t]);
      ck[t] = wmma_f32(a, bkf, ck[t]);
      cv[t] = wmma_f32(a, bvf, cv[t]);
    }
  }

  // C/D layout: vgpr g -> row (g + 8*hi), lane&15 -> col
  const float bqv = bq[h * 16 + col];
  const float bkv = bk[h * 16 + col];
  const float bvv = bv[h * 16 + col];
  float* tw = tb[h];

  #pragma unroll
  for (int t = 0; t < TT; ++t) {
    const int nodeT = node0 + t * 16;

    #pragma unroll
    for (int g = 0; g < 8; ++g)
      q[(nodeT + g + 8 * hi) * 128 + h * 16 + col] = cq[t][g] + bqv;

    // ---- k tile -> wave-private LDS -> per-relation 16x16 WMMA ----
    #pragma unroll
    for (int g = 0; g < 8; ++g)
      tw[(g + 8 * hi) * 17 + col] = ck[t][g] + bkv;
    #pragma unroll
    for (int r = 0; r < 2; ++r) {
      v8f c = {};
      const float* B = a_rel + ((r * 8 + h) * 16) * 16;  // [d][e] 16x16
      #pragma unroll
      for (int kk = 0; kk < 16; kk += 4) {
        const int ka = kk + 2 * hi;
        v2f a;
        a.x = tw[col * 17 + ka];
        a.y = tw[col * 17 + ka + 1];
        v2f b;
        b.x = B[ka * 16 + col];
        b.y = B[(ka + 1) * 16 + col];
        c = wmma_f32(a, b, c);
      }
      #pragma unroll
      for (int g = 0; g < 8; ++g)
        kr[r * NC + (nodeT + g + 8 * hi) * 128 + h * 16 + col] = c[g];
    }

    // ---- v tile -> LDS -> per-relation transform ----
    #pragma unroll
    for (int g = 0; g < 8; ++g)
      tw[(g + 8 * hi) * 17 + col] = cv[t][g] + bvv;
    #pragma unroll
    for (int r = 0; r < 2; ++r) {
      v8f c = {};
      const float* B = m_rel + ((r * 8 + h) * 16) * 16;
      #pragma unroll
      for (int kk = 0; kk < 16; kk += 4) {
        const int ka = kk + 2 * hi;
        v2f a;
        a.x = tw[col * 17 + ka];
        a.y = tw[col * 17 + ka + 1];
        v2f b;
        b.x = B[ka * 16 + col];
        b.y = B[(ka + 1) * 16 + col];
        c = wmma_f32(a, b, c);
      }
      #pragma unroll
      for (int g = 0; g < 8; ++g)
        vr[r * NC + (nodeT + g + 8 * hi) * 128 + h * 16 + col] = c[g];
    }
  }
}

// ---------------------------------------------------------------------------
// Kernel 2: per (edge,head) attention score + segment max (per dst,head,rel)
// ---------------------------------------------------------------------------
__global__ __launch_bounds__(256) void k_score(
    const int* __restrict__ ei, const int* __restrict__ et,
    const float* __restrict__ q, const float* __restrict__ kr,
    const float* __restrict__ p_rel, float* __restrict__ sc,
    float* __restrict__ mx) {
  const int idx = blockIdx.x * 256 + threadIdx.x;
  if (idx >= EH) return;
  const int e = idx >> 3;
  const int h = idx & 7;
  const int src = ei[e];
  const int dst = ei[EE + e];
  const int r = (et[e] == 1) ? 0 : 1;  // REL_EDGE_TYPE = (1,0)
  const float4* qd = (const float4*)(q + dst * 128 + h * 16);
  const float4* ks = (const float4*)(kr + r * NC + src * 128 + h * 16);
  float s = 0.0f;
  #pragma unroll
  for (int j = 0; j < 4; ++j) {
    const float4 a = qd[j];
    const float4 b = ks[j];
    s += a.x * b.x + a.y * b.y + a.z * b.z + a.w * b.w;
  }
  s *= p_rel[r * 8 + h] * 0.25f;  // scale = 1/sqrt(16)
  sc[idx] = s;
  atomicMax(&mx[(dst * 8 + h) * 2 + r], s);  // f32 max atomic
}

// ---------------------------------------------------------------------------
// Kernel 3: e = exp(score - max); den += e
// ---------------------------------------------------------------------------
__global__ __launch_bounds__(256) void k_expsum(const int* __restrict__ ei,
                                                const int* __restrict__ et,
                                                float* __restrict__ sc,
                                                const float* __restrict__ mx,
                                                float* __restrict__ den) {
  const int idx = blockIdx.x * 256 + threadIdx.x;
  if (idx >= EH) return;
  const int e = idx >> 3;
  const int h = idx & 7;
  const int dst = ei[EE + e];
  const int r = (et[e] == 1) ? 0 : 1;
  const float m = mx[(dst * 8 + h) * 2 + r];
  const float ev = expf(sc[idx] - m);
  sc[idx] = ev;
  atomicAdd(&den[(dst * 8 + h) * 2 + r], ev);
}

// ---------------------------------------------------------------------------
// Kernel 4: alpha = e/den ; attn[dst] += alpha * vr[src]
// ---------------------------------------------------------------------------
__global__ __launch_bounds__(256) void k_scatter(
    const int* __restrict__ ei, const int* __restrict__ et,
    const float* __restrict__ sc, const float* __restrict__ den,
    const float* __restrict__ vr, float* __restrict__ attn) {
  const int idx = blockIdx.x * 256 + threadIdx.x;
  if (idx >= EH) return;
  const int e = idx >> 3;
  const int h = idx & 7;
  const int src = ei[e];
  const int dst = ei[EE + e];
  const int r = (et[e] == 1) ? 0 : 1;
  const float dn = den[(dst * 8 + h) * 2 + r];
  const float alpha = sc[idx] / (dn > 0.0f ? dn : 1.0f);
  const float4* vs = (const float4*)(vr + r * NC + src * 128 + h * 16);
  float* ap = attn + dst * 128 + h * 16;
  #pragma unroll
  for (int j = 0; j < 4; ++j) {
    const float4 v = vs[j];
    atomicAdd(ap + j * 4 + 0, v.x * alpha);
    atomicAdd(ap + j * 4 + 1, v.y * alpha);
    atomicAdd(ap + j * 4 + 2, v.z * alpha);
    atomicAdd(ap + j * 4 + 3, v.w * alpha);
  }
}

// ---------------------------------------------------------------------------
// Kernel 5: out = sigmoid(skip)*(gelu(attn)@Wa + ba) + (1-sigmoid(skip))*x
// 80-node blocks, 5 M-tiles per wave (Wa fragments reused 5x).
// ---------------------------------------------------------------------------
__global__ __launch_bounds__(256) void k_out(const float* __restrict__ attn,
                                             const float* __restrict__ Wa,
                                             const float* __restrict__ ba,
                                             const float* __restrict__ x,
                                             const float* __restrict__ skip,
                                             float* __restrict__ out) {
  __shared__ float gs[TNODES * 128];  // 40 KB
  const int tid = threadIdx.x;
  const int lane = tid & 31;
  const int h = tid >> 5;
  const int node0 = blockIdx.x * TNODES;

  // gelu(attn tile) -> LDS (exact erf formulation, approximate=False)
  for (int i = tid; i < TNODES * 128; i += 256) {
    const float v = attn[node0 * 128 + i];
    gs[i] = 0.5f * v * (1.0f + erff(v * 0.70710678118654752f));
  }
  __syncthreads();

  const int col = lane & 15;
  const int hi = lane >> 4;
  v8f c[TT] = {};
  for (int k0 = 0; k0 < 128; k0 += 4) {
    const int ka = k0 + 2 * hi;
    v2f b;
    b.x = Wa[ka * 128 + h * 16 + col];
    b.y = Wa[(ka + 1) * 128 + h * 16 + col];
    #pragma unroll
    for (int t = 0; t < TT; ++t) {
      v2f a;
      a.x = gs[(t * 16 + col) * 128 + ka];
      a.y = gs[(t * 16 + col) * 128 + ka + 1];
      c[t] = wmma_f32(a, b, c[t]);
    }
  }

  const float aS = 1.0f / (1.0f + expf(-skip[0]));
  const float bav = ba[h * 16 + col];
  #pragma unroll
  for (int t = 0; t < TT; ++t) {
    #pragma unroll
    for (int g = 0; g < 8; ++g) {
      const int gi = (node0 + t * 16 + g + 8 * hi) * 128 + h * 16 + col;
      out[gi] = aS * (c[t][g] + bav) + (1.0f - aS) * x[gi];
    }
  }
}

// ---------------------------------------------------------------------------
extern "C" void kernel_launch(void* const* d_in, const int* in_sizes, int n_in,
                              void* d_out, int out_size, void* d_ws,
                              size_t ws_size, hipStream_t stream) {
  const float* x = (const float*)d_in[0];
  const int* ei = (const int*)d_in[1];      // [2,E]
  const int* et = (const int*)d_in[2];      // [E]
  const float* Wk = (const float*)d_in[3];
  const float* bk = (const float*)d_in[4];
  const float* Wq = (const float*)d_in[5];
  const float* bq = (const float*)d_in[6];
  const float* Wv = (const float*)d_in[7];
  const float* bv = (const float*)d_in[8];
  const float* a_rel = (const float*)d_in[9];   // [2,H,D,D]
  const float* m_rel = (const float*)d_in[10];  // [2,H,D,D]
  const float* p_rel = (const float*)d_in[11];  // [2,H]
  const float* Wa = (const float*)d_in[12];
  const float* ba = (const float*)d_in[13];
  const float* skip = (const float*)d_in[14];
  float* out = (float*)d_out;

  // scratch layout (floats)
  float* q = (float*)d_ws;       // N*C
  float* kr = q + NC;            // 2*N*C
  float* vr = kr + 2 * NC;       // 2*N*C
  float* attn = vr + 2 * NC;     // N*C
  float* mx = attn + NC;         // N*H*2
  float* den = mx + NH2;         // N*H*2
  float* sc = den + NH2;         // E*H

  k_init<<<2048, 256, 0, stream>>>(attn, mx, den);
  k_qkv<<<NN / TNODES, 256, 0, stream>>>(x, Wk, bk, Wq, bq, Wv, bv, a_rel,
                                         m_rel, q, kr, vr);
  const int eb = (EH + 255) / 256;
  k_score<<<eb, 256, 0, stream>>>(ei, et, q, kr, p_rel, sc, mx);
  k_expsum<<<eb, 256, 0, stream>>>(ei, et, sc, mx, den);
  k_scatter<<<eb, 256, 0, stream>>>(ei, et, sc, den, vr, attn);
  k_out<<<NN / TNODES, 256, 0, stream>>>(attn, Wa, ba, x, skip, out);
}